// HGATConv_22282290332325
// MI455X (gfx1250) — compile-verified
//
#include <hip/hip_runtime.h>
#include <hip/hip_bf16.h>

typedef __attribute__((ext_vector_type(2))) float v2f;
typedef __attribute__((ext_vector_type(4))) float v4f;
typedef __attribute__((ext_vector_type(8))) float v8f;
typedef __attribute__((ext_vector_type(2))) int   v2i;

#define N_NODES 4096
#define F_IN    512
#define F_OUT   512
#define HEADS   4
#define HEAD_DIM 128
#define ALPHA   0.2f
#define NEG_INF (-9.0e15f)

// ---------------------------------------------------------------------------
// Kernel 1: h = x @ W   (4096x512 @ 512x512, fp32, V_WMMA_F32_16X16X4_F32)
// Block = 4 waves; block tile 64x64; wave computes a 16x64 strip.
// ---------------------------------------------------------------------------
__global__ __launch_bounds__(128) void gemm_xw(const float* __restrict__ x,
                                               const float* __restrict__ W,
                                               float* __restrict__ h) {
  const int lane = threadIdx.x & 31;
  const int wave = threadIdx.x >> 5;
  const int half = lane >> 4;       // 0: K={0,1}, 1: K={2,3}
  const int lid  = lane & 15;
  const int kb   = half * 2;

  const int bm = blockIdx.x;        // 64-row tile
  const int bn = blockIdx.y;        // 64-col tile
  const int row = bm * 64 + wave * 16 + lid;

  v8f acc[4];
#pragma unroll
  for (int t = 0; t < 4; ++t) acc[t] = {};

  const float* xrow = x + (size_t)row * F_IN;
  for (int k = 0; k < F_IN; k += 4) {
    v2f a;
    a.x = xrow[k + kb + 0];
    a.y = xrow[k + kb + 1];
#pragma unroll
    for (int t = 0; t < 4; ++t) {
      const int col = bn * 64 + t * 16 + lid;
      v2f b;
      b.x = W[(size_t)(k + kb + 0) * F_OUT + col];
      b.y = W[(size_t)(k + kb + 1) * F_OUT + col];
      acc[t] = __builtin_amdgcn_wmma_f32_16x16x4_f32(
          false, a, false, b, (short)0, acc[t], false, false);
    }
  }

#pragma unroll
  for (int t = 0; t < 4; ++t) {
#pragma unroll
    for (int v = 0; v < 8; ++v) {
      const int m = bm * 64 + wave * 16 + v + half * 8;
      const int c = bn * 64 + t * 16 + lid;
      h[(size_t)m * F_OUT + c] = acc[t][v];
    }
  }
}

// ---------------------------------------------------------------------------
// Kernel 2: si[head][n] = h[n, head*128:+128] . a1 ; sj likewise with a2.
// Also writes sjT[n][head] (float4 per node) for the row-max scan.
// One wave per node.
// ---------------------------------------------------------------------------
__global__ __launch_bounds__(128) void attn_scores(const float* __restrict__ h,
                                                   const float* __restrict__ a,
                                                   float* __restrict__ si,
                                                   float* __restrict__ sj,
                                                   float* __restrict__ sjT) {
  const int lane = threadIdx.x & 31;
  const int wave = threadIdx.x >> 5;
  const int n = blockIdx.x * 4 + wave;

  const v4f a1 = *(const v4f*)(a + lane * 4);
  const v4f a2 = *(const v4f*)(a + HEAD_DIM + lane * 4);

#pragma unroll
  for (int head = 0; head < HEADS; ++head) {
    const float* hrow = h + (size_t)n * F_OUT + head * HEAD_DIM;
    const v4f hv = *(const v4f*)(hrow + lane * 4);
    float s1 = hv.x * a1.x + hv.y * a1.y + hv.z * a1.z + hv.w * a1.w;
    float s2 = hv.x * a2.x + hv.y * a2.y + hv.z * a2.z + hv.w * a2.w;
#pragma unroll
    for (int off = 16; off > 0; off >>= 1) {
      s1 += __shfl_down(s1, off, 32);
      s2 += __shfl_down(s2, off, 32);
    }
    if (lane == 0) {
      si[head * N_NODES + n] = s1;
      sj[head * N_NODES + n] = s2;
      sjT[n * HEADS + head] = s2;
    }
  }
}

// ---------------------------------------------------------------------------
// Kernel 3: Mx[head][i] = max_{j: adj[i][j]>0} sj[head][j]
// One 256-thread block per row; single adj scan covers all 4 heads.
// ---------------------------------------------------------------------------
__global__ __launch_bounds__(256) void row_max(const int* __restrict__ adj,
                                               const float* __restrict__ sjT,
                                               float* __restrict__ Mx) {
  const int i = blockIdx.x;
  const int t = threadIdx.x;
  v4f m = {NEG_INF, NEG_INF, NEG_INF, NEG_INF};
  const int* arow = adj + (size_t)i * N_NODES;
  for (int j = t; j < N_NODES; j += 256) {
    if (arow[j] > 0) {
      const v4f sv = *(const v4f*)(sjT + j * 4);
      m.x = fmaxf(m.x, sv.x);
      m.y = fmaxf(m.y, sv.y);
      m.z = fmaxf(m.z, sv.z);
      m.w = fmaxf(m.w, sv.w);
    }
  }
  __shared__ v4f red[256];
  red[t] = m;
  __syncthreads();
  for (int s = 128; s > 0; s >>= 1) {
    if (t < s) {
      v4f o = red[t + s];
      v4f c = red[t];
      c.x = fmaxf(c.x, o.x);
      c.y = fmaxf(c.y, o.y);
      c.z = fmaxf(c.z, o.z);
      c.w = fmaxf(c.w, o.w);
      red[t] = c;
    }
    __syncthreads();
  }
  if (t < 4) Mx[t * N_NODES + i] = red[0][t];
}

// ---------------------------------------------------------------------------
// Kernel 4: softmax numerator + P@V per head over a j-slice.
//   m_i = lrelu(si[i] + Mx[i])  (exact row max by monotonicity of leakyrelu)
//   p   = adj ? exp(lrelu(si+sj) - m_i) : 0
//   acc += P @ V  (WMMA f32 16x16x4), ssum += p
// Block = 4 waves = 64 rows of one head; V chunks staged in LDS in a
// row-pair interleaved layout so each B-fragment is one ds_load_b64.
// PARTIAL=true : write un-normalized partials (j-split over gridDim.z).
// PARTIAL=false: gridDim.z==1, normalize + elu + write final output.
// ---------------------------------------------------------------------------
#define CHUNK_J 64
#define PAIRS   (CHUNK_J / 2)
#define PAIR_STRIDE 288  // floats; 256 data + 32 pad => adjacent pairs hit
                         // complementary 32-bank halves for b64 reads

template <bool PARTIAL>
__global__ __launch_bounds__(128) void attn_agg(const float* __restrict__ h,
                                                const int* __restrict__ adj,
                                                const float* __restrict__ si,
                                                const float* __restrict__ sj,
                                                const float* __restrict__ Mx,
                                                float* __restrict__ outp,   // out or pacc base
                                                float* __restrict__ pssum,  // partial sums (PARTIAL)
                                                int jspan) {
  __shared__ float Vp[PAIRS * PAIR_STRIDE];
  __shared__ float srow[4][16];

  const int lane = threadIdx.x & 31;
  const int wave = threadIdx.x >> 5;
  const int half = lane >> 4;
  const int lid  = lane & 15;
  const int kb   = half * 2;

  const int head = blockIdx.y;
  const int iblk = blockIdx.x * 64;
  const int z    = blockIdx.z;
  const int jstart = z * jspan;
  const int row  = iblk + wave * 16 + lid;  // A-fragment row (dup across halves)

  const float sir = si[head * N_NODES + row];
  const float mx0 = sir + Mx[head * N_NODES + row];
  const float mrow = (mx0 > 0.f) ? mx0 : ALPHA * mx0;

  v8f acc[8];
#pragma unroll
  for (int t = 0; t < 8; ++t) acc[t] = {};
  float ssum = 0.f;

  const float* sjh  = sj + head * N_NODES;
  const int*   arow = adj + (size_t)row * N_NODES;
  const float* Vg   = h + head * HEAD_DIM;  // V[j][d] = h[j*512 + head*128 + d]

  // Staging decomposition: 128 threads cover PAIRS(32) x 32-d strips.
  const int st_pair = threadIdx.x >> 2;        // 0..31
  const int st_d0   = (threadIdx.x & 3) * 32;  // 0,32,64,96

  for (int c = jstart; c < jstart + jspan; c += CHUNK_J) {
    // Stage V chunk, interleaving row pairs: Vp[pair][d][0..1] = {V[2p][d], V[2p+1][d]}
    {
      const float* src0 = Vg + (size_t)(c + st_pair * 2 + 0) * F_OUT + st_d0;
      const float* src1 = Vg + (size_t)(c + st_pair * 2 + 1) * F_OUT + st_d0;
      float* dst = Vp + st_pair * PAIR_STRIDE + st_d0 * 2;
#pragma unroll
      for (int q = 0; q < 8; ++q) {
        const v4f r0 = *(const v4f*)(src0 + q * 4);
        const v4f r1 = *(const v4f*)(src1 + q * 4);
        v4f w0 = {r0.x, r1.x, r0.y, r1.y};
        v4f w1 = {r0.z, r1.z, r0.w, r1.w};
        *(v4f*)(dst + q * 8 + 0) = w0;
        *(v4f*)(dst + q * 8 + 4) = w1;
      }
    }
    __syncthreads();

#pragma unroll
    for (int s4 = 0; s4 < CHUNK_J / 4; ++s4) {
      const int jg0 = c + s4 * 4 + kb;  // even -> 8B-aligned paired loads
      const v2f sjv = *(const v2f*)(sjh + jg0);
      const v2i av  = *(const v2i*)(arow + jg0);

      float e0 = sir + sjv.x; e0 = (e0 > 0.f) ? e0 : ALPHA * e0;
      float e1 = sir + sjv.y; e1 = (e1 > 0.f) ? e1 : ALPHA * e1;
      const float p0 = (av.x > 0) ? __expf(e0 - mrow) : 0.f;
      const float p1 = (av.y > 0) ? __expf(e1 - mrow) : 0.f;
      ssum += p0 + p1;

      v2f af;
      af.x = p0;
      af.y = p1;
      // B rows (jl0, jl0+1) live interleaved in pair (s4*2 + half).
      const float* bp = Vp + (s4 * 2 + half) * PAIR_STRIDE + lid * 2;
#pragma unroll
      for (int t = 0; t < 8; ++t) {
        const v2f bf = *(const v2f*)(bp + t * 32);
        acc[t] = __builtin_amdgcn_wmma_f32_16x16x4_f32(
            false, af, false, bf, (short)0, acc[t], false, false);
      }
    }
    __syncthreads();
  }

  // Row numerator-sum: combine the two half-wave partials.
  ssum += __shfl_xor(ssum, 16, 32);

  if (PARTIAL) {
    if (half == 0) pssum[(size_t)z * (HEADS * N_NODES) + head * N_NODES + row] = ssum;
    float* pa = outp + (size_t)z * (N_NODES * F_OUT);
#pragma unroll
    for (int t = 0; t < 8; ++t) {
#pragma unroll
      for (int v = 0; v < 8; ++v) {
        const int m = iblk + wave * 16 + v + half * 8;
        pa[(size_t)m * F_OUT + head * HEAD_DIM + t * 16 + lid] = acc[t][v];
      }
    }
  } else {
    if (half == 0) srow[wave][lid] = ssum;
    __syncthreads();
#pragma unroll
    for (int t = 0; t < 8; ++t) {
#pragma unroll
      for (int v = 0; v < 8; ++v) {
        const int m = iblk + wave * 16 + v + half * 8;
        float val = acc[t][v] / srow[wave][v + half * 8];
        val = (val > 0.f) ? val : (__expf(val) - 1.f);  // elu
        outp[(size_t)m * F_OUT + head * HEAD_DIM + t * 16 + lid] = val;
      }
    }
  }
}

// ---------------------------------------------------------------------------
// Kernel 5 (j-split path): combine partials, normalize, elu.
// ---------------------------------------------------------------------------
template <int JS>
__global__ __launch_bounds__(256) void attn_reduce(const float* __restrict__ pacc,
                                                   const float* __restrict__ pssum,
                                                   float* __restrict__ out) {
  const int i = blockIdx.x;
#pragma unroll
  for (int r = 0; r < 2; ++r) {
    const int col = threadIdx.x + r * 256;
    const int head = col >> 7;
    float num = 0.f, den = 0.f;
#pragma unroll
    for (int zz = 0; zz < JS; ++zz) {
      num += pacc[(size_t)zz * (N_NODES * F_OUT) + (size_t)i * F_OUT + col];
      den += pssum[(size_t)zz * (HEADS * N_NODES) + head * N_NODES + i];
    }
    float val = num / den;
    val = (val > 0.f) ? val : (__expf(val) - 1.f);  // elu
    out[(size_t)i * F_OUT + col] = val;
  }
}

// ---------------------------------------------------------------------------
extern "C" void kernel_launch(void* const* d_in, const int* in_sizes, int n_in,
                              void* d_out, int out_size, void* d_ws, size_t ws_size,
                              hipStream_t stream) {
  const float* x   = (const float*)d_in[0];  // (4096, 512)
  const int*   adj = (const int*)d_in[1];    // (4096, 4096)
  const float* W   = (const float*)d_in[2];  // (512, 512)
  const float* a   = (const float*)d_in[3];  // (256, 1)
  float* out = (float*)d_out;                // (4096, 512)

  constexpr int JS = 4;
  const size_t HN = (size_t)HEADS * N_NODES;

  float* h   = (float*)d_ws;                  // 4096*512 f32 = 8 MB
  float* si  = h + (size_t)N_NODES * F_OUT;
  float* sj  = si + HN;
  float* sjT = sj + HN;
  float* Mx  = sjT + HN;
  float* pacc  = Mx + HN;                     // JS * 8 MB
  float* pssum = pacc + (size_t)JS * N_NODES * F_OUT;
  const size_t need_js =
      (size_t)(pssum + JS * HN - (float*)d_ws) * sizeof(float);

  gemm_xw<<<dim3(N_NODES / 64, F_OUT / 64), 128, 0, stream>>>(x, W, h);
  attn_scores<<<dim3(N_NODES / 4), 128, 0, stream>>>(h, a, si, sj, sjT);
  row_max<<<dim3(N_NODES), 256, 0, stream>>>(adj, sjT, Mx);

  if (ws_size >= need_js) {
    // j-split for occupancy: 64 x 4 x 4 = 1024 blocks (4096 waves).
    attn_agg<true><<<dim3(N_NODES / 64, HEADS, JS), 128, 0, stream>>>(
        h, adj, si, sj, Mx, pacc, pssum, N_NODES / JS);
    attn_reduce<JS><<<dim3(N_NODES), 256, 0, stream>>>(pacc, pssum, out);
  } else {
    attn_agg<false><<<dim3(N_NODES / 64, HEADS, 1), 128, 0, stream>>>(
        h, adj, si, sj, Mx, out, nullptr, N_NODES);
  }
}